// VectorQuantizer_9620726743262
// MI455X (gfx1250) — compile-verified
//
#include <hip/hip_runtime.h>
#include <math.h>

// Problem constants (fixed by reference)
#define NROWS   65536        // 64 * 32 * 32 flattened vectors
#define DIM     64           // embedding dim
#define KCB     512          // codebook entries
#define NELEM   4194304      // 64*64*32*32 elements of inputs / quantized
#define ENC_F2  16777216     // encodings as float2 count: 65536*512/2

typedef float v2f __attribute__((ext_vector_type(2)));
typedef float v8f __attribute__((ext_vector_type(8)));

// ---------------------------------------------------------------------------
// Init: zero histogram counts, precompute ||e_k||^2
// ---------------------------------------------------------------------------
__global__ __launch_bounds__(512)
void vq_init(const float* __restrict__ cb, int* __restrict__ counts,
             float* __restrict__ cbnorm) {
    int t = threadIdx.x;
    if (t < KCB) {
        counts[t] = 0;
        const float* r = cb + t * DIM;
        float s = 0.f;
#pragma unroll 8
        for (int j = 0; j < DIM; ++j) s += r[j] * r[j];
        cbnorm[t] = s;
    }
}

// ---------------------------------------------------------------------------
// Argmin via V_WMMA_F32_16X16X4_F32.
// Block = 256 threads = 8 waves; each wave owns a 16-row M-tile.
// score(n,k) = ||e_k||^2 - 2 * <x_n, e_k>   (||x||^2 constant per row)
// Two codebook tiles processed concurrently -> two independent WMMA
// accumulator chains + doubled B-load MLP.
// ---------------------------------------------------------------------------
__global__ __launch_bounds__(256)
void vq_argmin(const float* __restrict__ x, const float* __restrict__ cb,
               const float* __restrict__ cbnorm, int* __restrict__ idx_out,
               int* __restrict__ counts) {
    __shared__ int hist[KCB];
    const int tid = threadIdx.x;
    for (int i = tid; i < KCB; i += 256) hist[i] = 0;
    __syncthreads();

    const int lane = tid & 31;
    const int wave = tid >> 5;
    const int n0   = blockIdx.x * 128 + wave * 16;   // first flat row of tile
    const int m    = lane & 15;                      // row (A) / col (B/C) in tile
    const int cSel = (lane >> 4) << 1;               // K sub-select: 0 or 2
    const int b    = n0 >> 10;                       // 1024 rows per batch image
    const int hw   = n0 & 1023;
    // inputs layout [B, C, HW]: addr = b*65536 + c*1024 + hw
    const float* xb = x + (size_t)b * 65536 + hw + m;

    // Preload the full 16x64 A tile: 16 k-steps of K=4.
    // ISA f32 A 16x4 layout: lanes0-15 K={0,1}, lanes16-31 K={2,3}.
    v2f a[16];
#pragma unroll
    for (int kk = 0; kk < 16; ++kk) {
        a[kk].x = xb[(size_t)(4 * kk + cSel)     * 1024];
        a[kk].y = xb[(size_t)(4 * kk + cSel + 1) * 1024];
    }

    float bestv[8];
    int   besti[8];
#pragma unroll
    for (int v = 0; v < 8; ++v) { bestv[v] = 3.402823466e38f; besti[v] = 0; }

    // 16 iterations x 2 codebook tiles of 16 entries each
    for (int e0 = 0; e0 < KCB; e0 += 32) {
        const int col0 = e0 + m;                      // this lane's C column, tile 0
        const int col1 = col0 + 16;                   // tile 1
        const float* cbr0 = cb + (size_t)col0 * DIM + cSel;
        const float* cbr1 = cb + (size_t)col1 * DIM + cSel;
        v8f c0 = {0.f, 0.f, 0.f, 0.f, 0.f, 0.f, 0.f, 0.f};
        v8f c1 = {0.f, 0.f, 0.f, 0.f, 0.f, 0.f, 0.f, 0.f};
#pragma unroll
        for (int kk = 0; kk < 16; ++kk) {
            v2f b0, b1;
            b0.x = cbr0[4 * kk];
            b0.y = cbr0[4 * kk + 1];
            b1.x = cbr1[4 * kk];
            b1.y = cbr1[4 * kk + 1];
            c0 = __builtin_amdgcn_wmma_f32_16x16x4_f32(
                     false, a[kk], false, b0, (short)0, c0, false, false);
            c1 = __builtin_amdgcn_wmma_f32_16x16x4_f32(
                     false, a[kk], false, b1, (short)0, c1, false, false);
        }
        const float nrm0 = cbnorm[col0];
        const float nrm1 = cbnorm[col1];
#pragma unroll
        for (int v = 0; v < 8; ++v) {
            float s0 = nrm0 - 2.0f * c0[v];
            if (s0 < bestv[v]) { bestv[v] = s0; besti[v] = col0; }
            float s1 = nrm1 - 2.0f * c1[v];
            if (s1 < bestv[v]) { bestv[v] = s1; besti[v] = col1; }
        }
    }

    // Cross-lane min over the 16 columns held in each half-wave.
    // C layout: lanes0-15 rows M=v, lanes16-31 rows M=v+8.
#pragma unroll
    for (int v = 0; v < 8; ++v) {
        float val = bestv[v];
        int   ix  = besti[v];
#pragma unroll
        for (int off = 1; off < 16; off <<= 1) {
            float ov = __shfl_xor(val, off, 32);
            int   oi = __shfl_xor(ix,  off, 32);
            if (ov < val || (ov == val && oi < ix)) { val = ov; ix = oi; }
        }
        if (m == 0) {
            int row = n0 + v + ((lane >> 4) << 3);
            idx_out[row] = ix;
            atomicAdd(&hist[ix], 1);
        }
    }

    __syncthreads();
    for (int i = tid; i < KCB; i += 256) {
        int h = hist[i];
        if (h) atomicAdd(&counts[i], h);
    }
}

// ---------------------------------------------------------------------------
// Gather quantized values [B,C,H,W] + deterministic per-block MSE partials
// ---------------------------------------------------------------------------
__global__ __launch_bounds__(256)
void vq_quantize(const float* __restrict__ x, const float* __restrict__ cb,
                 const int* __restrict__ idx, float* __restrict__ outq,
                 float* __restrict__ partial) {
    const int t  = blockIdx.x * 256 + threadIdx.x;
    const int b  = t >> 16;
    const int c  = (t >> 10) & 63;
    const int hw = t & 1023;
    const int n  = (b << 10) | hw;
    const int k  = idx[n];
    const float q = cb[k * DIM + c];
    outq[t] = q;
    float d = q - x[t];
    float s = d * d;
#pragma unroll
    for (int off = 16; off; off >>= 1) s += __shfl_xor(s, off, 32);
    __shared__ float wsum[8];
    const int lane = threadIdx.x & 31, wave = threadIdx.x >> 5;
    if (lane == 0) wsum[wave] = s;
    __syncthreads();
    if (threadIdx.x == 0) {
        float tot = 0.f;
#pragma unroll
        for (int i = 0; i < 8; ++i) tot += wsum[i];
        partial[blockIdx.x] = tot;
    }
}

// ---------------------------------------------------------------------------
// One-hot encodings [N, 512]; float2 stores (segment is 8B-aligned only)
// ---------------------------------------------------------------------------
__global__ __launch_bounds__(256)
void vq_encodings(const int* __restrict__ idx, float2* __restrict__ enc) {
    const int stride = gridDim.x * blockDim.x;
    for (int i = blockIdx.x * blockDim.x + threadIdx.x; i < ENC_F2; i += stride) {
        const int row = i >> 8;             // 256 float2 per row of 512
        const int c0  = (i & 255) << 1;
        const int k   = idx[row];
        float2 v;
        v.x = (k == c0)     ? 1.0f : 0.0f;
        v.y = (k == c0 + 1) ? 1.0f : 0.0f;
        enc[i] = v;
    }
}

// ---------------------------------------------------------------------------
// Finalize: perplexity from counts, loss from MSE partials
// loss = q_latent + 0.25*e_latent = 1.25 * mean((q - x)^2)   (forward values)
// ---------------------------------------------------------------------------
__global__ __launch_bounds__(512)
void vq_finalize(const int* __restrict__ counts, const float* __restrict__ partial,
                 int nPartial, float* __restrict__ out_loss,
                 float* __restrict__ out_perp) {
    __shared__ float se[512];
    __shared__ float sl[512];
    const int t = threadIdx.x;
    const float p = (float)counts[t] * (1.0f / 65536.0f);
    se[t] = p * logf(p + 1e-10f);
    float ls = 0.f;
    for (int i = t; i < nPartial; i += 512) ls += partial[i];
    sl[t] = ls;
    __syncthreads();
    for (int s = 256; s; s >>= 1) {
        if (t < s) { se[t] += se[t + s]; sl[t] += sl[t + s]; }
        __syncthreads();
    }
    if (t == 0) {
        *out_perp = expf(-se[0]);
        *out_loss = sl[0] * (1.25f / 4194304.0f);
    }
}

// ---------------------------------------------------------------------------
extern "C" void kernel_launch(void* const* d_in, const int* in_sizes, int n_in,
                              void* d_out, int out_size, void* d_ws, size_t ws_size,
                              hipStream_t stream) {
    const float* x  = (const float*)d_in[0];   // [64, 64, 32, 32]
    const float* cb = (const float*)d_in[1];   // [512, 64]
    float* out = (float*)d_out;

    char* ws = (char*)d_ws;
    int*   idx     = (int*)  ws;                               // 65536 ints
    int*   counts  = (int*)  (ws + NROWS * 4);                 // 512 ints
    float* cbnorm  = (float*)(ws + NROWS * 4 + KCB * 4);       // 512 floats
    float* partial = (float*)(ws + NROWS * 4 + 2 * KCB * 4);   // 16384 floats

    float* out_loss = out;                 // [0]
    float* out_q    = out + 1;             // [1 .. 1+NELEM)
    float* out_perp = out + 1 + NELEM;     // [1+NELEM]
    float* out_enc  = out + 2 + NELEM;     // [2+NELEM ..), 8B-aligned

    vq_init     <<<1,            512, 0, stream>>>(cb, counts, cbnorm);
    vq_argmin   <<<NROWS / 128,  256, 0, stream>>>(x, cb, cbnorm, idx, counts);
    vq_quantize <<<NELEM / 256,  256, 0, stream>>>(x, cb, idx, out_q, partial);
    vq_encodings<<<8192,         256, 0, stream>>>(idx, (float2*)out_enc);
    vq_finalize <<<1,            512, 0, stream>>>(counts, partial, NELEM / 256,
                                                   out_loss, out_perp);
}